// GCN_RegressionModel_46316927320529
// MI455X (gfx1250) — compile-verified
//
#include <hip/hip_runtime.h>
#include <stdint.h>

typedef float v2f __attribute__((ext_vector_type(2)));
typedef float v4f __attribute__((ext_vector_type(4)));
typedef float v8f __attribute__((ext_vector_type(8)));

// ---------- degree / normalization ----------
__global__ void k_zero(float* p, int n) {
  int i = blockIdx.x * blockDim.x + threadIdx.x;
  if (i < n) p[i] = 0.0f;
}

__global__ void k_count_deg(const long long* __restrict__ dst, int E, float* __restrict__ deg) {
  int e = blockIdx.x * blockDim.x + threadIdx.x;
  if (e < E) atomicAdd(&deg[(int)dst[e]], 1.0f);
}

__global__ void k_dinv(float* deg_dinv, int N) {
  int i = blockIdx.x * blockDim.x + threadIdx.x;
  if (i < N) deg_dinv[i] = rsqrtf(deg_dinv[i] + 1.0f);  // +1 self-loop => always > 0
}

// ---------- pre-pack W1 into WMMA B-fragment layout (zero-padded to 16 cols) ----------
// frag1[step*32 + lane] = { W1[k+k0][n], W1[k+k0+1][n] }, k=4*step, k0=(lane/16)*2, n=lane%16
__global__ void k_fragB1(const float* __restrict__ W1, v2f* __restrict__ frag1) {
  int t = blockIdx.x * blockDim.x + threadIdx.x;   // 32 steps * 32 lanes = 1024
  if (t < 32 * 32) {
    int s = t >> 5, l = t & 31;
    int n = l & 15;
    int k = (s << 2) + ((l >> 4) << 1);
    v2f v;
    v.x = (n < 8) ? W1[k * 8 + n] : 0.0f;
    v.y = (n < 8) ? W1[(k + 1) * 8 + n] : 0.0f;
    frag1[t] = v;
  }
}

// ---------- GEMM1: xw1[N,8] = x[N,128] @ W1[128,8], f32 WMMA 16x16x4 ----------
__global__ void __launch_bounds__(256) k_gemm1_wmma(const float* __restrict__ x,
                                                    const v2f* __restrict__ frag1,
                                                    float* __restrict__ xw1, int N) {
  __shared__ float smem[8 * 16 * 128];             // 8 waves * 8KB tile
  int wave = threadIdx.x >> 5;
  int lane = threadIdx.x & 31;
  int tile = blockIdx.x * (blockDim.x >> 5) + wave;
  int base = tile * 16;
  float* sm = smem + wave * (16 * 128);
  // stage 16x128 x-tile: one fully-coalesced 512B row per pass
  if (base < N) {
    for (int r = 0; r < 16; ++r) {
      int gr = base + r; if (gr >= N) gr = N - 1;
      *(v4f*)(sm + r * 128 + lane * 4) = *(const v4f*)(x + (size_t)gr * 128 + lane * 4);
    }
  }
  __syncthreads();
  if (base >= N) return;                 // wave-uniform: EXEC all-ones below
  int m  = lane & 15;                    // A row / B,D column for this lane
  int k0 = (lane >> 4) << 1;             // 0 or 2 (lane-half K offset)
  const float* arow = sm + m * 128;
  v8f d = {0.f, 0.f, 0.f, 0.f, 0.f, 0.f, 0.f, 0.f};
  #pragma unroll 4
  for (int s = 0; s < 32; ++s) {         // K = 4*s
    v2f a = *(const v2f*)(arow + (s << 2) + k0);   // ds_load_b64
    v2f b = frag1[(s << 5) + lane];                // coalesced global_load_b64
    d = __builtin_amdgcn_wmma_f32_16x16x4_f32(false, a, false, b, (short)0, d, false, false);
  }
  if (m < 8) {                           // only first 8 columns are real
    int rbase = base + ((lane >> 4) << 3);
    float* o = xw1 + (size_t)rbase * 8 + m;
    if (base + 16 <= N) {                // full tile: unguarded, clause-able stores
      #pragma unroll
      for (int v = 0; v < 8; ++v) o[(size_t)v * 8] = d[v];
    } else {                             // boundary tile (not hit when N%16==0)
      #pragma unroll
      for (int v = 0; v < 8; ++v)
        if (rbase + v < N) o[(size_t)v * 8] = d[v];
    }
  }
}

// ---------- self-loop init: out8[i] = dinv[i]^2 * feat8[i] ----------
__global__ void k_selfloop_init(const float* __restrict__ feat8, const float* __restrict__ dinv,
                                float* __restrict__ out8, int N) {
  int i = blockIdx.x * blockDim.x + threadIdx.x;
  if (i < N) {
    float w = dinv[i] * dinv[i];
    #pragma unroll
    for (int c = 0; c < 8; ++c) out8[(size_t)i * 8 + c] = w * feat8[(size_t)i * 8 + c];
  }
}

// ---------- 8-wide edge scatter: out8[dst] += dinv[s]*dinv[d]*feat8[src] ----------
__global__ void k_scatter8(const long long* __restrict__ ei, int E, const float* __restrict__ dinv,
                           const float* __restrict__ feat8, float* __restrict__ out8) {
  int e = blockIdx.x * blockDim.x + threadIdx.x;
  if (e < E) {
    int s = (int)ei[e];
    int d = (int)ei[(size_t)E + e];
    float w = dinv[s] * dinv[d];
    const float* f = feat8 + (size_t)s * 8;
    float* o = out8 + (size_t)d * 8;
    #pragma unroll
    for (int c = 0; c < 8; ++c) atomicAdd(&o[c], w * f[c]);
  }
}

// ---------- h1 = relu(agg1 + b1) in place; agg2_init = dinv^2 * h1 ----------
__global__ void k_bias_relu_init2(float* __restrict__ agg1_h1, const float* __restrict__ b1,
                                  const float* __restrict__ dinv, float* __restrict__ agg2, int N) {
  int i = blockIdx.x * blockDim.x + threadIdx.x;
  if (i < N) {
    float w = dinv[i] * dinv[i];
    #pragma unroll
    for (int c = 0; c < 8; ++c) {
      float h = fmaxf(agg1_h1[(size_t)i * 8 + c] + b1[c], 0.0f);
      agg1_h1[(size_t)i * 8 + c] = h;     // h1 stored in place
      agg2[(size_t)i * 8 + c] = w * h;    // self-loop term of layer 2
    }
  }
}

// ---------- fold tail weights: W2p[8,256] = W2@Wl1 ; bp = b2@Wl1 + bl1 ----------
__global__ void k_fold(const float* __restrict__ W2, const float* __restrict__ b2,
                       const float* __restrict__ Wl1, const float* __restrict__ bl1,
                       float* __restrict__ W2p, float* __restrict__ bp) {
  int c = threadIdx.x;  // 256 threads, 1 block
  for (int k = 0; k < 8; ++k) {
    float s = 0.0f;
    for (int j = 0; j < 256; ++j) s += W2[k * 256 + j] * Wl1[j * 256 + c];
    W2p[k * 256 + c] = s;
  }
  float s = bl1[c];
  for (int j = 0; j < 256; ++j) s += b2[j] * Wl1[j * 256 + c];
  bp[c] = s;
}

// ---------- pre-pack W2p into WMMA B-fragment layout ----------
// frag2[(chunk*2+step)*32 + lane] = { W2p[k][c], W2p[k+1][c] },
//   c = chunk*16 + lane%16, k = step*4 + (lane/16)*2
__global__ void k_fragB2(const float* __restrict__ W2p, v2f* __restrict__ frag2) {
  int t = blockIdx.x * blockDim.x + threadIdx.x;   // 16 chunks * 2 steps * 32 lanes = 1024
  if (t < 16 * 2 * 32) {
    int l = t & 31;
    int step = (t >> 5) & 1;
    int chunk = t >> 6;
    int c = (chunk << 4) + (l & 15);
    int k = (step << 2) + ((l >> 4) << 1);
    v2f v;
    v.x = W2p[k * 256 + c];
    v.y = W2p[(k + 1) * 256 + c];
    frag2[t] = v;
  }
}

// ---------- fused tail: out[i] = relu(agg2[i,:8] @ W2p + bp) . Wl2 + bl2 ----------
__global__ void __launch_bounds__(256) k_out_wmma(const float* __restrict__ agg2,
                                                  const v2f* __restrict__ frag2,
                                                  const float* __restrict__ bp,
                                                  const float* __restrict__ Wl2,
                                                  const float* __restrict__ bl2,
                                                  float* __restrict__ out, int N) {
  int wave = threadIdx.x >> 5;
  int lane = threadIdx.x & 31;
  int tile = blockIdx.x * (blockDim.x >> 5) + wave;
  int base = tile * 16;
  if (base >= N) return;                 // wave-uniform
  int m  = lane & 15;
  int k0 = (lane >> 4) << 1;             // 0 or 2
  int row = base + m; if (row >= N) row = N - 1;
  const float* ar = agg2 + (size_t)row * 8;
  v2f a0 = *(const v2f*)(ar + k0);       // K = 0..3 step (b64 load)
  v2f a1 = *(const v2f*)(ar + 4 + k0);   // K = 4..7 step
  float acc[8];
  #pragma unroll
  for (int v = 0; v < 8; ++v) acc[v] = 0.0f;

  #pragma unroll 4
  for (int ch = 0; ch < 16; ++ch) {      // 16 column chunks of W2p
    int c = (ch << 4) + m;
    v2f b0 = frag2[((ch << 1) + 0) * 32 + lane];   // coalesced b64
    v2f b1 = frag2[((ch << 1) + 1) * 32 + lane];
    v8f d = {0.f, 0.f, 0.f, 0.f, 0.f, 0.f, 0.f, 0.f};
    d = __builtin_amdgcn_wmma_f32_16x16x4_f32(false, a0, false, b0, (short)0, d, false, false);
    d = __builtin_amdgcn_wmma_f32_16x16x4_f32(false, a1, false, b1, (short)0, d, false, false);
    float bias = bp[c];
    float wl   = Wl2[c];
    #pragma unroll
    for (int v = 0; v < 8; ++v) acc[v] += fmaxf(d[v] + bias, 0.0f) * wl;
  }
  // reduce across the 16 lanes of each half-wave (columns of the D layout)
  #pragma unroll
  for (int mask = 1; mask <= 8; mask <<= 1) {
    #pragma unroll
    for (int v = 0; v < 8; ++v) acc[v] += __shfl_xor(acc[v], mask, 32);
  }
  if (m == 0) {                                // lanes 0 and 16 hold row sums
    int rbase = base + ((lane >> 4) << 3);
    float bl = bl2[0];
    float* o = out + rbase;
    if (base + 16 <= N) {                      // full tile: unguarded stores
      #pragma unroll
      for (int v = 0; v < 8; ++v) o[v] = acc[v] + bl;
    } else {
      #pragma unroll
      for (int v = 0; v < 8; ++v)
        if (rbase + v < N) o[v] = acc[v] + bl;
    }
  }
}

extern "C" void kernel_launch(void* const* d_in, const int* in_sizes, int n_in,
                              void* d_out, int out_size, void* d_ws, size_t ws_size,
                              hipStream_t stream) {
  const float*     x   = (const float*)d_in[0];
  const long long* ei  = (const long long*)d_in[1];   // int64 [2, E]
  const float*     W1  = (const float*)d_in[2];
  const float*     b1  = (const float*)d_in[3];
  const float*     W2  = (const float*)d_in[4];
  const float*     b2  = (const float*)d_in[5];
  const float*     Wl1 = (const float*)d_in[6];
  const float*     bl1 = (const float*)d_in[7];
  const float*     Wl2 = (const float*)d_in[8];
  const float*     bl2 = (const float*)d_in[9];
  float* out = (float*)d_out;

  int N = in_sizes[0] / 128;
  int E = in_sizes[1] / 2;

  float* ws    = (float*)d_ws;
  float* dinv  = ws;                         // N
  float* xw1   = dinv + N;                   // N*8
  float* agg1  = xw1 + (size_t)N * 8;        // N*8  (becomes h1 in place)
  float* agg2  = agg1 + (size_t)N * 8;       // N*8
  float* W2p   = agg2 + (size_t)N * 8;       // 8*256
  float* bp    = W2p + 8 * 256;              // 256
  v2f*   frag1 = (v2f*)(bp + 256);           // 1024 v2f (8KB)
  v2f*   frag2 = frag1 + 1024;               // 1024 v2f (8KB)

  const int TB = 256;
  int nB = (N + TB - 1) / TB;
  int eB = (E + TB - 1) / TB;
  int tiles = (N + 15) / 16;
  int wpb   = TB / 32;                      // 8 waves per block
  int tB    = (tiles + wpb - 1) / wpb;

  k_zero<<<nB, TB, 0, stream>>>(dinv, N);
  k_count_deg<<<eB, TB, 0, stream>>>(ei + E, E, dinv);
  k_dinv<<<nB, TB, 0, stream>>>(dinv, N);

  k_fragB1<<<4, TB, 0, stream>>>(W1, frag1);
  k_gemm1_wmma<<<tB, TB, 0, stream>>>(x, frag1, xw1, N);

  k_selfloop_init<<<nB, TB, 0, stream>>>(xw1, dinv, agg1, N);
  k_scatter8<<<eB, TB, 0, stream>>>(ei, E, dinv, xw1, agg1);
  k_bias_relu_init2<<<nB, TB, 0, stream>>>(agg1, b1, dinv, agg2, N);
  k_scatter8<<<eB, TB, 0, stream>>>(ei, E, dinv, agg1 /*h1*/, agg2);

  k_fold<<<1, 256, 0, stream>>>(W2, b2, Wl1, bl1, W2p, bp);
  k_fragB2<<<4, TB, 0, stream>>>(W2p, frag2);
  k_out_wmma<<<tB, TB, 0, stream>>>(agg2, frag2, bp, Wl2, bl2, out, N);
}